// AttentionalSampler_29059748725602
// MI455X (gfx1250) — compile-verified
//
#include <hip/hip_runtime.h>
#include <hip/hip_bf16.h>

typedef __attribute__((ext_vector_type(2))) float v2f;
typedef __attribute__((ext_vector_type(8))) float v8f;

#define BTOT   128      // B*T
#define MTOK   64       // tokens per (b,t)
#define DDIM   128      // feature dim
#define PPAT   1024     // patches per (b,t) (32x32)
#define WSTR   132      // padded LDS row stride for 128-wide tiles (bank-conflict free)
#define SSTR   1028     // padded LDS row stride for 1024-wide score rows
#define SCALE  0.08838834764831845f   // 1/sqrt(128)
#define NEGC   -0.14391156828f        // -4*ln(100)/128

__device__ __forceinline__ v8f wmma_f32(v2f a, v2f b, v8f c) {
  // D = A(16x4) * B(4x16) + C, fp32 WMMA
  return __builtin_amdgcn_wmma_f32_16x16x4_f32(
      /*neg_a=*/false, a, /*neg_b=*/false, b,
      /*c_mod=*/(short)0, c, /*reuse_a=*/false, /*reuse_b=*/false);
}

// RoPE one half-row (64 channels = 16 groups of 4), in-register; accumulates
// sum / sumsq for the subsequent LayerNorm.
__device__ __forceinline__ void rope_half(const float* src, float* vals, int half,
                                          float h, float w, float& sum, float& sumsq) {
  sum = 0.f; sumsq = 0.f;
  #pragma unroll
  for (int g = 0; g < 16; ++g) {
    int gg = half * 16 + g;                       // global group 0..31
    float theta = __expf(NEGC * (float)(gg + 1)); // 100^(-4*(g+1)/128)
    float sh, ch, sw, cw;
    __sincosf(theta * h, &sh, &ch);
    __sincosf(theta * w, &sw, &cw);
    float a = src[4*g+0], c = src[4*g+1], b = src[4*g+2], e = src[4*g+3];
    float a2 = a * ch - b * sh;
    float b2 = b * ch - a2 * sh;   // torch in-place semantics: uses updated a2
    float c2 = c * cw + e * sw;
    float e2 = e * cw + c2 * sw;   // uses updated c2
    vals[4*g+0] = a2; vals[4*g+1] = c2; vals[4*g+2] = b2; vals[4*g+3] = e2;
    sum   += a2 + c2 + b2 + e2;
    sumsq += a2*a2 + c2*c2 + b2*b2 + e2*e2;
  }
}

// ---------------------------------------------------------------------------
// Kernel 1: K = LN(RoPE(mv @ Wk^T + bk)).  grid = BTOT*8, block = 256 (8 waves)
// Each block: 128 patch rows of one (b,t).  Wk staged in LDS.
// ---------------------------------------------------------------------------
__global__ __launch_bounds__(256) void k_proj_kernel(
    const float* __restrict__ mv, const float* __restrict__ Wk,
    const float* __restrict__ bk, const float* __restrict__ ln_g,
    const float* __restrict__ ln_b, float* __restrict__ k_ln) {
  extern __shared__ float smem[];
  float* Wk_s = smem;                 // 128 * WSTR
  float* kt   = smem + 128 * WSTR;    // 128 * WSTR (result tile, pre-LN)

  const int tid = threadIdx.x;
  const int bt  = blockIdx.x >> 3;
  const int rg  = blockIdx.x & 7;     // row-group: 128 patches each

  for (int idx = tid; idx < DDIM * DDIM; idx += 256)
    Wk_s[(idx >> 7) * WSTR + (idx & 127)] = Wk[idx];
  __syncthreads();

  const int wave = tid >> 5, lane = tid & 31, lo = lane & 15, hi = lane >> 4;
  const int row0 = rg * 128 + wave * 16;
  const float* arow = mv + ((size_t)bt * PPAT + row0 + lo) * DDIM;

  v8f zero = {};
  v8f acc[8];
  #pragma unroll
  for (int ct = 0; ct < 8; ++ct) acc[ct] = zero;

  for (int kk = 0; kk < 32; ++kk) {
    int d0 = 4 * kk + 2 * hi;
    v2f a = *(const v2f*)(arow + d0);
    #pragma unroll
    for (int ct = 0; ct < 8; ++ct) {
      v2f b = *(const v2f*)&Wk_s[(ct * 16 + lo) * WSTR + d0];
      acc[ct] = wmma_f32(a, b, acc[ct]);
    }
  }
  #pragma unroll
  for (int ct = 0; ct < 8; ++ct) {
    float bias = bk[ct * 16 + lo];
    #pragma unroll
    for (int r = 0; r < 8; ++r)
      kt[(wave * 16 + r + 8 * hi) * WSTR + ct * 16 + lo] = acc[ct][r] + bias;
  }
  __syncthreads();

  // RoPE + LayerNorm: 2 threads per row, half-row each, in registers.
  const int row = tid >> 1, half = tid & 1;
  const int p = rg * 128 + row;
  float h = (float)(p >> 5), w = (float)(p & 31);
  float vals[64], sum, sumsq;
  rope_half(&kt[row * WSTR + half * 64], vals, half, h, w, sum, sumsq);
  sum   += __shfl_xor(sum, 1);
  sumsq += __shfl_xor(sumsq, 1);
  float mean = sum * (1.0f / 128.0f);
  float var  = sumsq * (1.0f / 128.0f) - mean * mean;
  float rstd = rsqrtf(var + 1e-5f);
  float* dst = k_ln + ((size_t)bt * PPAT + p) * DDIM + half * 64;
  #pragma unroll
  for (int i = 0; i < 64; ++i) {
    int ch = half * 64 + i;
    dst[i] = (vals[i] - mean) * rstd * ln_g[ch] + ln_b[ch];
  }
}

// ---------------------------------------------------------------------------
// Kernel 2: Q = LN(RoPE(t @ Wq^T + bq)).  grid = BTOT, block = 128 (4 waves)
// ---------------------------------------------------------------------------
__global__ __launch_bounds__(128) void q_proj_kernel(
    const float* __restrict__ tin, const long long* __restrict__ positions,
    const float* __restrict__ Wq, const float* __restrict__ bq,
    const float* __restrict__ ln_g, const float* __restrict__ ln_b,
    float* __restrict__ q_ln) {
  extern __shared__ float smem[];
  float* Wq_s = smem;                 // 128 * WSTR
  float* qt   = smem + 128 * WSTR;    // 64 * WSTR

  const int tid = threadIdx.x;
  const int bt  = blockIdx.x;

  for (int idx = tid; idx < DDIM * DDIM; idx += 128)
    Wq_s[(idx >> 7) * WSTR + (idx & 127)] = Wq[idx];
  __syncthreads();

  const int wave = tid >> 5, lane = tid & 31, lo = lane & 15, hi = lane >> 4;
  const float* arow = tin + ((size_t)bt * MTOK + wave * 16 + lo) * DDIM;

  v8f zero = {};
  v8f acc[8];
  #pragma unroll
  for (int ct = 0; ct < 8; ++ct) acc[ct] = zero;

  for (int kk = 0; kk < 32; ++kk) {
    int d0 = 4 * kk + 2 * hi;
    v2f a = *(const v2f*)(arow + d0);
    #pragma unroll
    for (int ct = 0; ct < 8; ++ct) {
      v2f b = *(const v2f*)&Wq_s[(ct * 16 + lo) * WSTR + d0];
      acc[ct] = wmma_f32(a, b, acc[ct]);
    }
  }
  #pragma unroll
  for (int ct = 0; ct < 8; ++ct) {
    float bias = bq[ct * 16 + lo];
    #pragma unroll
    for (int r = 0; r < 8; ++r)
      qt[(wave * 16 + r + 8 * hi) * WSTR + ct * 16 + lo] = acc[ct][r] + bias;
  }
  __syncthreads();

  const int row = tid >> 1, half = tid & 1;
  long long pv = positions[(size_t)bt * MTOK + row];
  float h = (float)(int)(pv >> 5), w = (float)(int)(pv & 31);
  float vals[64], sum, sumsq;
  rope_half(&qt[row * WSTR + half * 64], vals, half, h, w, sum, sumsq);
  sum   += __shfl_xor(sum, 1);
  sumsq += __shfl_xor(sumsq, 1);
  float mean = sum * (1.0f / 128.0f);
  float var  = sumsq * (1.0f / 128.0f) - mean * mean;
  float rstd = rsqrtf(var + 1e-5f);
  float* dst = q_ln + ((size_t)bt * MTOK + row) * DDIM + half * 64;
  #pragma unroll
  for (int i = 0; i < 64; ++i) {
    int ch = half * 64 + i;
    dst[i] = (vals[i] - mean) * rstd * ln_g[ch] + ln_b[ch];
  }
}

// ---------------------------------------------------------------------------
// Kernel 3: fused attention per (b,t).  grid = BTOT, block = 256 (8 waves).
// Full 64x1024 score matrix lives in LDS (CDNA5: 320 KB/WGP).
// ---------------------------------------------------------------------------
__global__ __launch_bounds__(256) void attn_kernel(
    const float* __restrict__ q_ln, const float* __restrict__ k_ln,
    const float* __restrict__ mv, const long long* __restrict__ positions,
    float* __restrict__ out) {
  extern __shared__ float smem[];
  float* q_s  = smem;                         // 64 * 132  = 8448
  float* S    = smem + 64 * WSTR;             // 64 * 1028 = 65792
  float* rsum = S + 64 * SSTR;                // 64
  float* phw  = rsum + 64;                    // 128 (ph[64], pw[64])

  const int tid = threadIdx.x;
  const int bt  = blockIdx.x;
  const int wave = tid >> 5, lane = tid & 31, lo = lane & 15, hi = lane >> 4;

  for (int idx = tid; idx < MTOK * DDIM; idx += 256) {
    int r = idx >> 7, c = idx & 127;
    q_s[r * WSTR + c] = q_ln[((size_t)bt * MTOK + r) * DDIM + c];
  }
  if (tid < MTOK) {
    long long pv = positions[(size_t)bt * MTOK + tid];
    phw[tid]      = (float)(int)(pv >> 5);
    phw[64 + tid] = (float)(int)(pv & 31);
  }
  __syncthreads();

  // ---- Stage 1: S = Q @ K^T * scale + distance bias -----------------------
  const float* kbase = k_ln + (size_t)bt * PPAT * DDIM;
  for (int cti = 0; cti < 8; ++cti) {
    int pcol = (wave * 8 + cti) * 16 + lo;          // this lane's patch column
    const float* krow = kbase + (size_t)pcol * DDIM;
    v8f zero = {};
    v8f acc[4];
    #pragma unroll
    for (int rt = 0; rt < 4; ++rt) acc[rt] = zero;
    for (int kk = 0; kk < 32; ++kk) {
      int d0 = 4 * kk + 2 * hi;
      v2f b = *(const v2f*)(krow + d0);
      #pragma unroll
      for (int rt = 0; rt < 4; ++rt) {
        v2f a = *(const v2f*)&q_s[(rt * 16 + lo) * WSTR + d0];
        acc[rt] = wmma_f32(a, b, acc[rt]);
      }
    }
    float pc_h = (float)(pcol >> 5), pc_w = (float)(pcol & 31);
    #pragma unroll
    for (int rt = 0; rt < 4; ++rt) {
      #pragma unroll
      for (int r = 0; r < 8; ++r) {
        int m = rt * 16 + r + 8 * hi;
        float dh = phw[m] - pc_h, dw = phw[64 + m] - pc_w;
        float bias = -sqrtf(dh * dh + dw * dw) * 0.125f;  // -dist/(2*DECAY^2)
        S[m * SSTR + pcol] = acc[rt][r] * SCALE + bias;
      }
    }
  }
  __syncthreads();

  // ---- Stage 2: softmax over 1024 cols, 4 threads/row ---------------------
  {
    int row = tid >> 2, seg = tid & 3;
    float* Srow = S + row * SSTR;
    float mx = -1e30f;
    for (int i = 0; i < 256; ++i) mx = fmaxf(mx, Srow[seg + 4 * i]);
    mx = fmaxf(mx, __shfl_xor(mx, 1));
    mx = fmaxf(mx, __shfl_xor(mx, 2));
    float s = 0.f;
    for (int i = 0; i < 256; ++i) {
      int c = seg + 4 * i;
      float e = __expf(Srow[c] - mx);
      Srow[c] = e;
      s += e;
    }
    s += __shfl_xor(s, 1);
    s += __shfl_xor(s, 2);
    if (seg == 0) rsum[row] = s;
  }
  __syncthreads();

  // ---- Stage 3: out = (exp-scores @ V) / rowsum ---------------------------
  const int rt = wave >> 1;
  const int ctbase = (wave & 1) * 4;
  const float* vbase = mv + (size_t)bt * PPAT * DDIM;
  v8f zero = {};
  v8f acc[4];
  #pragma unroll
  for (int c = 0; c < 4; ++c) acc[c] = zero;
  for (int s3 = 0; s3 < 256; ++s3) {
    int p0 = 4 * s3 + 2 * hi;
    v2f a = *(const v2f*)&S[(rt * 16 + lo) * SSTR + p0];
    #pragma unroll
    for (int cti = 0; cti < 4; ++cti) {
      int nn = (ctbase + cti) * 16 + lo;
      v2f b;
      b[0] = vbase[(size_t)p0 * DDIM + nn];
      b[1] = vbase[(size_t)(p0 + 1) * DDIM + nn];
      acc[cti] = wmma_f32(a, b, acc[cti]);
    }
  }
  #pragma unroll
  for (int cti = 0; cti < 4; ++cti) {
    #pragma unroll
    for (int r = 0; r < 8; ++r) {
      int m = rt * 16 + r + 8 * hi;
      out[((size_t)bt * MTOK + m) * DDIM + (ctbase + cti) * 16 + lo] =
          acc[cti][r] / rsum[m];
    }
  }
}

extern "C" void kernel_launch(void* const* d_in, const int* in_sizes, int n_in,
                              void* d_out, int out_size, void* d_ws, size_t ws_size,
                              hipStream_t stream) {
  (void)in_sizes; (void)n_in; (void)out_size; (void)ws_size;
  const float*     tin  = (const float*)d_in[0];
  const float*     mv   = (const float*)d_in[1];
  const long long* pos  = (const long long*)d_in[2];
  const float*     Wq   = (const float*)d_in[3];
  const float*     bq   = (const float*)d_in[4];
  const float*     Wk   = (const float*)d_in[5];
  const float*     bk   = (const float*)d_in[6];
  const float*     ln_g = (const float*)d_in[7];
  const float*     ln_b = (const float*)d_in[8];
  float* outp = (float*)d_out;

  float* k_ln = (float*)d_ws;                         // 128*1024*128 f32 = 64 MB
  float* q_ln = k_ln + (size_t)BTOT * PPAT * DDIM;    // 128*64*128  f32 = 4 MB

  size_t lds1 = (size_t)(128 * WSTR + 128 * WSTR) * sizeof(float);          // 132 KB
  size_t lds2 = (size_t)(128 * WSTR + 64 * WSTR) * sizeof(float);           // 99 KB
  size_t lds3 = (size_t)(64 * WSTR + 64 * SSTR + 64 + 128) * sizeof(float); // ~291 KB

  k_proj_kernel<<<dim3(BTOT * 8), dim3(256), lds1, stream>>>(mv, Wk, bk, ln_g, ln_b, k_ln);
  q_proj_kernel<<<dim3(BTOT), dim3(128), lds2, stream>>>(tin, pos, Wq, bq, ln_g, ln_b, q_ln);
  attn_kernel<<<dim3(BTOT), dim3(256), lds3, stream>>>(q_ln, k_ln, mv, pos, outp);
}